// I2P_88983132438581
// MI455X (gfx1250) — compile-verified
//
#include <hip/hip_runtime.h>
#include <hip/hip_bf16.h>
#include <math.h>

typedef float v2f __attribute__((ext_vector_type(2)));
typedef float v8f __attribute__((ext_vector_type(8)));

#define HW_   32400   // 180*180
#define NPTS_ 162000  // 5*HW
#define C_    128

// ---------------------------------------------------------------------------
// Fold weights:  Wqk[k,j] = sum_i Wq[i,k]*Wk[i,j]   (so qk = lidar^T @ Wqk + bqk)
//                bqk[j]   = sum_i bq[i]*Wk[i,j]
//                Wov[i,j] = sum_t Wo[i,t]*Wv[t,j]   (so out = ctx @ Wov^T + bov)
//                bov[i]   = sum_t Wo[i,t]*bv[t] + bo[i]
// bk is dropped: constant across the 5 keys -> softmax shift-invariant.
// ---------------------------------------------------------------------------
__global__ void fold_weights_kernel(const float* __restrict__ Wq, const float* __restrict__ bq,
                                    const float* __restrict__ Wk,
                                    const float* __restrict__ Wv, const float* __restrict__ bv,
                                    const float* __restrict__ Wo, const float* __restrict__ bo,
                                    float* __restrict__ Wqk, float* __restrict__ bqk,
                                    float* __restrict__ Wov, float* __restrict__ bov) {
    int i = blockIdx.x;   // 0..127
    int j = threadIdx.x;  // 0..127
    float sqk = 0.f, sov = 0.f;
    for (int t = 0; t < C_; ++t) {
        sqk += Wq[t * C_ + i] * Wk[t * C_ + j];
        sov += Wo[i * C_ + t] * Wv[t * C_ + j];
    }
    Wqk[i * C_ + j] = sqk;
    Wov[i * C_ + j] = sov;
    if (i == 0) {
        float s = 0.f;
        for (int t = 0; t < C_; ++t) s += bq[t] * Wk[t * C_ + j];
        bqk[j] = s;
    }
    if (j == 0) {
        float s = 0.f;
        for (int t = 0; t < C_; ++t) s += Wo[i * C_ + t] * bv[t];
        bov[i] = s + bo[i];
    }
}

// ---------------------------------------------------------------------------
// Project each voxel center into the 6 cameras and bilinearly sample the
// 128-channel image features; average over valid cameras.
// One 128-thread block per point; lanes 0..5 compute per-camera projection
// into LDS, then each thread handles one channel.
// ---------------------------------------------------------------------------
__global__ void project_sample_kernel(const float* __restrict__ img,   // [6,128,32,88]
                                      const float* __restrict__ l2i,   // [6,4,4]
                                      float* __restrict__ reduced,     // [N,128]
                                      float* __restrict__ cnt) {       // [N]
    __shared__ float sx[6], sy[6];
    __shared__ int   sok[6];
    const int n   = blockIdx.x;
    const int z   = n / HW_;
    const int rem = n - z * HW_;
    const int iy  = rem / 180;
    const int ix  = rem - iy * 180;

    if (threadIdx.x < 6) {
        const int cam = threadIdx.x;
        const float px = ((float)ix + 0.5f) * 0.6f - 54.f;   // [-54,54] over 180
        const float py = ((float)iy + 0.5f) * 0.6f - 54.f;
        const float pz = ((float)z  + 0.5f) * 1.6f - 5.f;    // [-5,3]  over 5
        const float* M = l2i + cam * 16;
        const float X  = M[0]*px + M[1]*py + M[2]*pz  + M[3];
        const float Y  = M[4]*px + M[5]*py + M[6]*pz  + M[7];
        const float Dd = M[8]*px + M[9]*py + M[10]*pz + M[11];
        bool ok = Dd > 1e-5f;
        const float inv = 1.f / fmaxf(Dd, 1e-5f);
        const float gx = (X * inv * (1.f/704.f) - 0.5f) * 2.f;
        const float gy = (Y * inv * (1.f/256.f) - 0.5f) * 2.f;
        ok = ok && (gx > -1.f) && (gx < 1.f) && (gy > -1.f) && (gy < 1.f);
        sx[cam]  = (gx + 1.f) * 44.f - 0.5f;  // 88-wide feature map
        sy[cam]  = (gy + 1.f) * 16.f - 0.5f;  // 32-high feature map
        sok[cam] = ok ? 1 : 0;
    }
    __syncthreads();

    const int c = threadIdx.x;
    float acc = 0.f, count = 0.f;
    #pragma unroll
    for (int cam = 0; cam < 6; ++cam) {
        if (sok[cam]) {
            count += 1.f;
            const float xs = sx[cam], ys = sy[cam];
            const float x0f = floorf(xs), y0f = floorf(ys);
            const int x0 = (int)x0f, y0 = (int)y0f;
            const float wx1 = xs - x0f, wy1 = ys - y0f;
            const float wx0 = 1.f - wx1, wy0 = 1.f - wy1;
            const float* base = img + (size_t)(cam * C_ + c) * (32 * 88);
            const bool xa = (x0 >= 0 && x0 <= 87),  xb = (x0+1 >= 0 && x0+1 <= 87);
            const bool ya = (y0 >= 0 && y0 <= 31),  yb = (y0+1 >= 0 && y0+1 <= 31);
            const float t00 = (xa && ya) ? base[y0*88 + x0]       : 0.f;
            const float t01 = (xb && ya) ? base[y0*88 + x0+1]     : 0.f;
            const float t10 = (xa && yb) ? base[(y0+1)*88 + x0]   : 0.f;
            const float t11 = (xb && yb) ? base[(y0+1)*88 + x0+1] : 0.f;
            acc += t00*(wx0*wy0) + t01*(wx1*wy0) + t10*(wx0*wy1) + t11*(wx1*wy1);
        }
    }
    reduced[(size_t)n * C_ + c] = acc / (count + 1e-10f);
    if (c == 0) cnt[n] = count;
}

// ---------------------------------------------------------------------------
// WMMA fp32 GEMM: D[M,128] = A[M,128] @ B + bias, K=N=128.
// TRANS_A: A[m,k] read as A[k*ldA + m] (lidar feature is channel-major).
// TRANS_B: B[k,n] = W[k*128+n] else W[n*128+k] (i.e. A @ W^T).
// TRANS_OUT: store D[n*M + m] scaled by mask[m] (final masked transpose).
// One wave owns a 16x128 slab: 32 K-steps x 8 N-tiles of V_WMMA_F32_16X16X4_F32.
// W (64 KB) staged in LDS once per 8-wave workgroup.
// ---------------------------------------------------------------------------
template <bool TRANS_A, bool TRANS_B, bool TRANS_OUT>
__global__ __launch_bounds__(256)
void gemm128_wmma(const float* __restrict__ A, const float* __restrict__ W,
                  const float* __restrict__ bias, const float* __restrict__ mask,
                  float* __restrict__ D, int M, int ldA) {
    __shared__ float sW[C_ * C_];
    for (int i = threadIdx.x; i < (C_ * C_) / 4; i += 256)
        ((float4*)sW)[i] = ((const float4*)W)[i];
    __syncthreads();

    const int wave  = threadIdx.x >> 5;
    const int lane  = threadIdx.x & 31;
    const int mtile = blockIdx.x * 8 + wave;
    if (mtile * 16 >= M) return;

    const int m     = mtile * 16 + (lane & 15);
    const int khalf = (lane >> 4) << 1;          // 0 or 2 per 16x4 A layout

    v8f acc[8] = {};
    for (int k0 = 0; k0 < C_; k0 += 4) {
        v2f a;
        if (TRANS_A) {
            a[0] = A[(size_t)(k0 + khalf)     * ldA + m];
            a[1] = A[(size_t)(k0 + khalf + 1) * ldA + m];
        } else {
            const float* ap = A + (size_t)m * C_ + k0 + khalf;
            a[0] = ap[0];
            a[1] = ap[1];
        }
        #pragma unroll
        for (int j = 0; j < 8; ++j) {
            const int n = j * 16 + (lane & 15);
            v2f b;
            if (TRANS_B) {
                b[0] = sW[(k0 + khalf)     * C_ + n];
                b[1] = sW[(k0 + khalf + 1) * C_ + n];
            } else {
                b[0] = sW[n * C_ + k0 + khalf];
                b[1] = sW[n * C_ + k0 + khalf + 1];
            }
            acc[j] = __builtin_amdgcn_wmma_f32_16x16x4_f32(
                false, a, false, b, (short)0, acc[j], false, false);
        }
    }

    const int mrow_off = (lane >> 4) << 3;       // C/D: lanes 16-31 hold M=8..15
    #pragma unroll
    for (int j = 0; j < 8; ++j) {
        const int n  = j * 16 + (lane & 15);
        const float bn = bias[n];
        #pragma unroll
        for (int r = 0; r < 8; ++r) {
            const int mr  = mtile * 16 + mrow_off + r;
            const float v = acc[j][r] + bn;
            if (TRANS_OUT) D[(size_t)n * M + mr] = v * mask[mr];
            else           D[(size_t)mr * C_ + n] = v;
        }
    }
}

// ---------------------------------------------------------------------------
// Per-BEV-cell attention over Z=5 keys. One wave per cell (wave32): each lane
// holds 4 channels; dot products reduced with __shfl_xor.
// ctx[hw] = sum_z softmax_z(qk[hw] . reduced[z]) * reduced[z]
// ---------------------------------------------------------------------------
__global__ __launch_bounds__(256)
void attn_kernel(const float* __restrict__ qk, const float* __restrict__ reduced,
                 const float* __restrict__ cnt, float* __restrict__ ctx,
                 float* __restrict__ validp) {
    const int wave = threadIdx.x >> 5;
    const int lane = threadIdx.x & 31;
    const int hw   = blockIdx.x * 8 + wave;
    if (hw >= HW_) return;
    const float scale = 0.08838834764831845f;  // 1/sqrt(128)

    const float4 qv = ((const float4*)(qk + (size_t)hw * C_))[lane];
    float4 rv[5];
    float  s[5];
    bool   km[5];
    #pragma unroll
    for (int z = 0; z < 5; ++z) {
        const int n = z * HW_ + hw;
        rv[z] = ((const float4*)(reduced + (size_t)n * C_))[lane];
        float p = qv.x*rv[z].x + qv.y*rv[z].y + qv.z*rv[z].z + qv.w*rv[z].w;
        p += __shfl_xor(p, 16, 32);
        p += __shfl_xor(p, 8, 32);
        p += __shfl_xor(p, 4, 32);
        p += __shfl_xor(p, 2, 32);
        p += __shfl_xor(p, 1, 32);
        s[z]  = p * scale;
        km[z] = cnt[n] > 0.5f;
    }
    float mx = -3.0e38f;
    #pragma unroll
    for (int z = 0; z < 5; ++z) if (km[z] && s[z] > mx) mx = s[z];
    const bool any = (mx > -2.9e38f);
    float w[5], denom = 0.f;
    #pragma unroll
    for (int z = 0; z < 5; ++z) { w[z] = (any && km[z]) ? __expf(s[z] - mx) : 0.f; denom += w[z]; }
    const float inv = any ? 1.f / denom : 0.f;
    float4 c = make_float4(0.f, 0.f, 0.f, 0.f);
    #pragma unroll
    for (int z = 0; z < 5; ++z) {
        const float a = w[z] * inv;
        c.x += a*rv[z].x; c.y += a*rv[z].y; c.z += a*rv[z].z; c.w += a*rv[z].w;
    }
    ((float4*)(ctx + (size_t)hw * C_))[lane] = c;
    if (lane == 0) validp[hw] = any ? 1.f : 0.f;
}

// ---------------------------------------------------------------------------
extern "C" void kernel_launch(void* const* d_in, const int* in_sizes, int n_in,
                              void* d_out, int out_size, void* d_ws, size_t ws_size,
                              hipStream_t stream) {
    const float* lf  = (const float*)d_in[0];   // [128, 32400] channel-major
    const float* img = (const float*)d_in[1];   // [6,128,32,88]
    const float* l2i = (const float*)d_in[2];   // [6,4,4]
    const float* Wq  = (const float*)d_in[3];
    const float* bq  = (const float*)d_in[4];
    const float* Wk  = (const float*)d_in[5];
    // d_in[6] = bk: dropped (softmax shift-invariant across Z)
    const float* Wv  = (const float*)d_in[7];
    const float* bv  = (const float*)d_in[8];
    const float* Wo  = (const float*)d_in[9];
    const float* bo  = (const float*)d_in[10];
    float* out = (float*)d_out;

    float* ws      = (float*)d_ws;
    float* reduced = ws;                                   // N*128
    float* cnt     = reduced + (size_t)NPTS_ * C_;         // N
    float* qk      = cnt + NPTS_;                          // HW*128
    float* ctx     = qk + (size_t)HW_ * C_;                // HW*128
    float* validp  = ctx + (size_t)HW_ * C_;               // HW
    float* pWqk    = validp + HW_;                         // 128*128
    float* pbqk    = pWqk + C_ * C_;                       // 128
    float* pWov    = pbqk + C_;                            // 128*128
    float* pbov    = pWov + C_ * C_;                       // 128

    fold_weights_kernel<<<C_, C_, 0, stream>>>(Wq, bq, Wk, Wv, bv, Wo, bo,
                                               pWqk, pbqk, pWov, pbov);
    project_sample_kernel<<<NPTS_, C_, 0, stream>>>(img, l2i, reduced, cnt);

    const int mtiles = HW_ / 16;                 // 2025
    const int gblocks = (mtiles + 7) / 8;        // 254
    // qk = lidar^T @ Wqk + bqk     (TRANS_A over channel-major lidar, TRANS_B)
    gemm128_wmma<true, true, false><<<gblocks, 256, 0, stream>>>(
        lf, pWqk, pbqk, nullptr, qk, HW_, HW_);

    attn_kernel<<<(HW_ + 7) / 8, 256, 0, stream>>>(qk, reduced, cnt, ctx, validp);

    // out = (ctx @ Wov^T + bov) * valid, stored transposed to [128,180,180]
    gemm128_wmma<false, false, true><<<gblocks, 256, 0, stream>>>(
        ctx, pWov, pbov, validp, out, HW_, C_);
}